// Model_33002528702887
// MI455X (gfx1250) — compile-verified
//
#include <hip/hip_runtime.h>
#include <hip/hip_bf16.h>
#include <cstdint>

typedef float v4f __attribute__((ext_vector_type(4)));

#define NUM_EXPERTS 256
#define N_GROUP 8
#define TOPK_GROUP 4
#define WAVES_PER_BLOCK 8
#define BLOCK_THREADS 256
#define MAX_BLOCKS 4096

// Issue an async DMA of this lane's 32-byte slice of a 1KB score row into LDS.
// Two b128 async-to-LDS ops; ASYNCcnt += 2. INST_OFFSET applies to both the
// global and the LDS address (CDNA5 ISA), so one base pair
// covers both halves.
__device__ __forceinline__ void async_prefetch_row(const float* gsrc, uint32_t lds_off) {
    asm volatile(
        "global_load_async_to_lds_b128 %0, %1, off\n\t"
        "global_load_async_to_lds_b128 %0, %1, off offset:16"
        :: "v"(lds_off), "v"(gsrc) : "memory");
}

__global__ __launch_bounds__(BLOCK_THREADS) void moe_group_topk_kernel(
    const float* __restrict__ scores,
    float* __restrict__ out_scores,
    float* __restrict__ out_mask,
    int num_tokens)
{
    __shared__ float smem[WAVES_PER_BLOCK][2][NUM_EXPERTS];

    const int lane   = threadIdx.x & 31;
    const int wib    = threadIdx.x >> 5;                     // wave in block
    const int nwaves = gridDim.x * WAVES_PER_BLOCK;
    const int wave_id = blockIdx.x * WAVES_PER_BLOCK + wib;
    const int g = lane >> 2;                                 // this lane's group (0..7)

    float* buf0 = &smem[wib][0][0];
    float* buf1 = &smem[wib][1][0];
    // Generic LDS pointer low 32 bits == allocation-relative LDS byte address.
    const uint32_t lds0 = (uint32_t)(uintptr_t)(buf0 + lane * 8);
    const uint32_t lds1 = (uint32_t)(uintptr_t)(buf1 + lane * 8);

    const float NEG_INF = -__builtin_huge_valf();

    int t = wave_id;
    if (t < num_tokens)
        async_prefetch_row(scores + (size_t)t * NUM_EXPERTS + lane * 8, lds0);

    int cur = 0;
    for (; t < num_tokens; t += nwaves) {
        const int nxt = t + nwaves;
        if (nxt < num_tokens) {
            // Prefetch next row into the other buffer, then wait for only the
            // oldest 2 async ops (current row). Async loads complete in order.
            async_prefetch_row(scores + (size_t)nxt * NUM_EXPERTS + lane * 8,
                               cur ? lds0 : lds1);
            asm volatile("s_wait_asynccnt 2" ::: "memory");
        } else {
            asm volatile("s_wait_asynccnt 0" ::: "memory");
        }

        const float* buf = cur ? buf1 : buf0;
        v4f a = *(const v4f*)(buf + lane * 8);       // ds_load_b128
        v4f b = *(const v4f*)(buf + lane * 8 + 4);   // ds_load_b128

        // ---- per-lane top-2 of its 8 values ----
        float m1 = a.x, m2 = NEG_INF;
#define TOP2_STEP(v) { float nm1 = fmaxf(m1, (v)); m2 = fmaxf(m2, fminf(m1, (v))); m1 = nm1; }
        TOP2_STEP(a.y) TOP2_STEP(a.z) TOP2_STEP(a.w)
        TOP2_STEP(b.x) TOP2_STEP(b.y) TOP2_STEP(b.z) TOP2_STEP(b.w)
#undef TOP2_STEP

        // ---- merge top-2 pairs across the quad (lanes 4g..4g+3) ----
#pragma unroll
        for (int k = 1; k <= 2; k <<= 1) {
            float o1 = __shfl_xor(m1, k, 32);
            float o2 = __shfl_xor(m2, k, 32);
            float nm1 = fmaxf(m1, o1);
            m2 = fmaxf(fminf(m1, o1), fmaxf(m2, o2));
            m1 = nm1;
        }
        const float gs = m1 + m2;   // group score, identical across the quad

        // ---- rank of this group among the 8 group scores ----
        // jax.lax.top_k tie-break: equal values -> lower index wins.
        // (g^d) < g  <=>  g has the highest set bit of d.
        int rank = 0;
#pragma unroll
        for (int d = 1; d < 8; ++d) {
            float o = __shfl_xor(gs, d << 2, 32);   // score of group g^d
            const int hb = (d >= 4) ? 4 : ((d >= 2) ? 2 : 1);
            const bool tie_lower = (g & hb) != 0;
            rank += (int)((o > gs) || ((o == gs) && tie_lower));
        }
        const bool sel = (rank < TOPK_GROUP);

        // ---- masked scores (write-once: non-temporal b128 stores) ----
        const v4f ninf4 = {NEG_INF, NEG_INF, NEG_INF, NEG_INF};
        v4f ra = sel ? a : ninf4;
        v4f rb = sel ? b : ninf4;
        float* op = out_scores + (size_t)t * NUM_EXPERTS + lane * 8;
        __builtin_nontemporal_store(ra, (v4f*)op);
        __builtin_nontemporal_store(rb, (v4f*)(op + 4));

        // ---- group mask: lane L<8 takes group L's selection from lane 4L ----
        int selL = __shfl((int)sel, lane * 4, 32);
        if (lane < N_GROUP)
            __builtin_nontemporal_store((float)selL,
                                        out_mask + (size_t)t * N_GROUP + lane);
        cur ^= 1;
    }
}

extern "C" void kernel_launch(void* const* d_in, const int* in_sizes, int n_in,
                              void* d_out, int out_size, void* d_ws, size_t ws_size,
                              hipStream_t stream) {
    const float* scores = (const float*)d_in[0];
    const int num_tokens = in_sizes[0] / NUM_EXPERTS;

    float* out = (float*)d_out;
    float* out_scores = out;                                        // N*256 floats
    float* out_mask   = out + (size_t)num_tokens * NUM_EXPERTS;     // N*8 floats

    // ~8 tokens per wave so the async double-buffer pipeline has depth.
    int waves_needed = (num_tokens + 7) / 8;
    int blocks = (waves_needed + WAVES_PER_BLOCK - 1) / WAVES_PER_BLOCK;
    if (blocks > MAX_BLOCKS) blocks = MAX_BLOCKS;
    if (blocks < 1) blocks = 1;

    moe_group_topk_kernel<<<blocks, BLOCK_THREADS, 0, stream>>>(
        scores, out_scores, out_mask, num_tokens);
}